// PriorBetaModule_83288005804662
// MI455X (gfx1250) — compile-verified
//
#include <hip/hip_runtime.h>

typedef float v2f __attribute__((ext_vector_type(2)));
typedef float v8f __attribute__((ext_vector_type(8)));

#define N_STOCKS 65536
#define HIDDEN 512
#define N_IND 128
#define NEG_SLOPE 0.01f

// ---------------------------------------------------------------- phase 0
__global__ __launch_bounds__(256) void k_zero(float* __restrict__ S, int* __restrict__ bcnt) {
    int t = blockIdx.x * blockDim.x + threadIdx.x;
    if (t < N_IND * HIDDEN) S[t] = 0.0f;
    if (t < N_IND) bcnt[t] = 0;
}

// ---------------------------------------------------------------- phase 1
// One wave32 per stock: lane l loads int4 at columns [4l, 4l+4) of the one-hot
// row (coalesced 512B per wave), the unique lane that finds the 1 records the
// industry and bumps the edge-degree counter.
__global__ __launch_bounds__(256) void k_ind(const int* __restrict__ im,
                                             int* __restrict__ ind,
                                             int* __restrict__ bcnt) {
    int gt = blockIdx.x * blockDim.x + threadIdx.x;
    int stock = gt >> 5;
    int lane = gt & 31;
    if (stock >= N_STOCKS) return;
    const int4 v = ((const int4*)(im + (size_t)stock * N_IND))[lane];
    int col = -1;
    if (v.x == 1) col = lane * 4 + 0;
    else if (v.y == 1) col = lane * 4 + 1;
    else if (v.z == 1) col = lane * 4 + 2;
    else if (v.w == 1) col = lane * 4 + 3;
    if (col >= 0) {
        ind[stock] = col;
        atomicAdd(&bcnt[col], 1);
    }
}

// ---------------------------------------------------------------- phase 2
// Segment-sum e_s into S[128][512]. Each block owns (stock chunk of 2048) x
// (64-column slice): accumulate into a 128x64 f32 LDS tile with ds_add_f32
// atomics, then flush with global f32 atomics (32 partials per S element).
__global__ __launch_bounds__(256) void k_aggregate(const float* __restrict__ es,
                                                   const int* __restrict__ ind,
                                                   float* __restrict__ S) {
    __shared__ float lds[N_IND * 64];
    const int hc = blockIdx.x & 7;   // 8 column slices of 64
    const int sc = blockIdx.x >> 3;  // 32 stock chunks of 2048
    for (int t = threadIdx.x; t < N_IND * 64; t += blockDim.x) lds[t] = 0.0f;
    __syncthreads();

    const int col = threadIdx.x & 63;   // column within slice
    const int srow = threadIdx.x >> 6;  // 4 stocks in flight
    const int base = sc * 2048;
    const float* esc = es + (size_t)hc * 64 + col;
    for (int i = base + srow; i < base + 2048; i += 4) {
        int e = ind[i];
        float v = esc[(size_t)i * HIDDEN];
        atomicAdd(&lds[e * 64 + col], v);
    }
    __syncthreads();

    for (int t = threadIdx.x; t < N_IND * 64; t += blockDim.x) {
        float p = lds[t];
        if (p != 0.0f) {
            int e = t >> 6, c = t & 63;
            atomicAdd(&S[(size_t)e * HIDDEN + hc * 64 + c], p);
        }
    }
}

// ---------------------------------------------------------------- phase 3
// edge[e][h] = (sum_k S[e][k] * W[k][h]) * Binv[e] + bias[h]
// 128x512x512 GEMM via V_WMMA_F32_16X16X4_F32: one wave per 16x16 tile,
// 8 (M) x 32 (N) = 256 waves, 128 k-steps of K=4.
__global__ __launch_bounds__(256) void k_gemm(const float* __restrict__ S,
                                              const float* __restrict__ W,
                                              const int* __restrict__ bcnt,
                                              const float* __restrict__ bias,
                                              float* __restrict__ edge) {
    const int wave = (blockIdx.x * blockDim.x + threadIdx.x) >> 5;  // 0..255
    const int lane = threadIdx.x & 31;
    const int mt = wave >> 5;  // 0..7
    const int nt = wave & 31;  // 0..31
    const int m0 = mt * 16, n0 = nt * 16;
    const int l16 = lane & 15;
    const int hi = lane >> 4;  // 0: K=k,k+1 ; 1: K=k+2,k+3

    v8f acc = {};
    const float* arow = S + (size_t)(m0 + l16) * HIDDEN + hi * 2;  // A: lane=M
    const float* bcol = W + (size_t)(hi * 2) * HIDDEN + n0 + l16;  // B: lane=N

#pragma unroll 4
    for (int k = 0; k < HIDDEN; k += 4) {
        v2f a = *(const v2f*)(arow + k);  // S[m][k+2hi], S[m][k+2hi+1]
        v2f b;
        b.x = bcol[(size_t)k * HIDDEN];           // W[k+2hi][n]
        b.y = bcol[(size_t)k * HIDDEN + HIDDEN];  // W[k+2hi+1][n]
        acc = __builtin_amdgcn_wmma_f32_16x16x4_f32(
            /*neg_a=*/false, a, /*neg_b=*/false, b,
            /*c_mod=*/(short)0, acc, /*reuse_a=*/false, /*reuse_b=*/false);
    }

    const float bs = bias[n0 + l16];
#pragma unroll
    for (int r = 0; r < 8; ++r) {
        int m = m0 + r + hi * 8;  // C/D layout: VGPR r, lanes>=16 hold M=r+8
        int c = bcnt[m];
        float binv = c > 0 ? 1.0f / (float)c : 0.0f;
        edge[(size_t)m * HIDDEN + n0 + l16] = acc[r] * binv + bs;
    }
}

// ---------------------------------------------------------------- phase 4
// out[i][:] = LeakyReLU(edge[ind[i]][:]) ; float4 (B128) streams, edge in L2.
__global__ __launch_bounds__(256) void k_scatter(const float* __restrict__ edge,
                                                 const int* __restrict__ ind,
                                                 float* __restrict__ out) {
    size_t t = (size_t)blockIdx.x * blockDim.x + threadIdx.x;
    if (t >= (size_t)N_STOCKS * (HIDDEN / 4)) return;
    int i = (int)(t >> 7);       // stock
    int c4 = (int)(t & 127);     // float4 column
    int e = ind[i];
    float4 v = ((const float4*)edge)[(size_t)e * (HIDDEN / 4) + c4];
    v.x = v.x >= 0.0f ? v.x : NEG_SLOPE * v.x;
    v.y = v.y >= 0.0f ? v.y : NEG_SLOPE * v.y;
    v.z = v.z >= 0.0f ? v.z : NEG_SLOPE * v.z;
    v.w = v.w >= 0.0f ? v.w : NEG_SLOPE * v.w;
    ((float4*)out)[t] = v;
}

extern "C" void kernel_launch(void* const* d_in, const int* in_sizes, int n_in,
                              void* d_out, int out_size, void* d_ws, size_t ws_size,
                              hipStream_t stream) {
    const float* e_s = (const float*)d_in[0];
    const int* im = (const int*)d_in[1];
    const float* W = (const float*)d_in[2];
    const float* bias = (const float*)d_in[3];
    float* out = (float*)d_out;

    char* ws = (char*)d_ws;
    int* ind = (int*)ws;                              // 65536 * 4   = 262144 B
    int* bcnt = (int*)(ws + 262144);                  // 128 * 4     = 512 B
    float* S = (float*)(ws + 262144 + 512);           // 128*512*4   = 262144 B
    float* edge = (float*)(ws + 262144 + 512 + 262144);  // 262144 B

    // phase 0: zero accumulators (ws is not re-poisoned between replays)
    k_zero<<<(N_IND * HIDDEN + 255) / 256, 256, 0, stream>>>(S, bcnt);
    // phase 1: industry index + edge degree (one wave per stock)
    k_ind<<<(N_STOCKS * 32) / 256, 256, 0, stream>>>(im, ind, bcnt);
    // phase 2: segment sums (8 column slices x 32 stock chunks)
    k_aggregate<<<8 * 32, 256, 0, stream>>>(e_s, ind, S);
    // phase 3: WMMA GEMM + Binv + bias (256 waves = 32 blocks x 8 waves)
    k_gemm<<<32, 256, 0, stream>>>(S, W, bcnt, bias, edge);
    // phase 4: scatter + LeakyReLU
    k_scatter<<<(N_STOCKS * (HIDDEN / 4)) / 256, 256, 0, stream>>>(edge, ind, out);
}